// GraphLayer_10788957848285
// MI455X (gfx1250) — compile-verified
//
#include <hip/hip_runtime.h>
#include <hip/hip_bf16.h>
#include <stddef.h>

typedef __attribute__((ext_vector_type(2))) float v2f;
typedef __attribute__((ext_vector_type(8))) float v8f;

#define D  128
#define ED 32

// ---------- helpers ----------
__device__ __forceinline__ float wave_sum(float v) {
#pragma unroll
  for (int off = 16; off > 0; off >>= 1) v += __shfl_xor(v, off, 32);
  return v;
}

__device__ __forceinline__ unsigned f32_ord(float f) {
  unsigned b = __float_as_uint(f);
  return (b & 0x80000000u) ? ~b : (b | 0x80000000u);
}
__device__ __forceinline__ float ord_f32(unsigned o) {
  unsigned b = (o & 0x80000000u) ? (o ^ 0x80000000u) : ~o;
  return __uint_as_float(b);
}

// ---------- generic zero ----------
__global__ void k_zero(float* p, size_t n) {
  size_t i = (size_t)blockIdx.x * blockDim.x + threadIdx.x;
  size_t stride = (size_t)gridDim.x * blockDim.x;
  for (; i < n; i += stride) p[i] = 0.0f;
}

// ---------- ce = We @ a_e  (both layers; ce[0..31]=L1, ce[32..63]=L2) ----------
__global__ void k_prep_ce(const float* __restrict__ We1, const float* __restrict__ ae1,
                          const float* __restrict__ We2, const float* __restrict__ ae2,
                          float* __restrict__ ce) {
  int t = threadIdx.x;                 // 0..63
  int j = t & 31, which = t >> 5;
  const float* We = which ? We2 : We1;
  const float* ae = which ? ae2 : ae1;
  float s = 0.f;
  for (int n = 0; n < D; ++n) s += We[j * D + n] * ae[n];
  ce[t] = s;
}

// ---------- per-edge attr dot (both layers in one pass over edge_attr) ----------
__global__ void k_edge_pre(const float* __restrict__ ea, const float* __restrict__ ce,
                           float* __restrict__ eE1, float* __restrict__ eE2, int E) {
  int e = blockIdx.x * (blockDim.x >> 5) + (threadIdx.x >> 5);
  if (e >= E) return;
  int lane = threadIdx.x & 31;
  float v = ea[(size_t)e * ED + lane];
  float s1 = wave_sum(v * ce[lane]);
  float s2 = wave_sum(v * ce[32 + lane]);
  if (lane == 0) { eE1[e] = s1; eE2[e] = s2; }
}

// ---------- Y[M x 128] = X[M x 128] @ W[128 x 128] via v_wmma_f32_16x16x4_f32 ----------
// grid.x = M/16 (M must be multiple of 16; 50000 is), block = 256 (8 waves),
// wave w owns output n-tile w. A-tile staged in LDS (pad 132 to dodge bank conflicts).
__global__ void k_gemm_xW(const float* __restrict__ X, const float* __restrict__ W,
                          float* __restrict__ Y) {
  __shared__ float tileX[16 * 132];
  const int m0 = blockIdx.x * 16;
  // stage 16x128 A tile, coalesced
#pragma unroll
  for (int i = 0; i < 8; ++i) {
    int idx = threadIdx.x + i * 256;        // 0..2047
    int r = idx >> 7, c = idx & 127;
    tileX[r * 132 + c] = X[(size_t)(m0 + r) * D + c];
  }
  __syncthreads();

  const int wave = threadIdx.x >> 5;
  const int lane = threadIdx.x & 31;
  const int hi = lane >> 4;                 // lane half: 0 or 1
  const int lo = lane & 15;
  const int n0 = wave * 16;

  v8f c = {};
  for (int k = 0; k < D; k += 4) {
    int ka = k + hi * 2;
    // A 16x4 f32: lanes0-15 row=lo K=ka,ka+1 ; lanes16-31 row=lo K=ka(+2 offset via hi)
    v2f a, b;
    a.x = tileX[lo * 132 + ka];
    a.y = tileX[lo * 132 + ka + 1];
    // B 4x16 f32: VGPR0 -> K=ka, N=n0+lo ; VGPR1 -> K=ka+1
    b.x = W[(size_t)ka * D + n0 + lo];
    b.y = W[(size_t)(ka + 1) * D + n0 + lo];
    c = __builtin_amdgcn_wmma_f32_16x16x4_f32(false, a, false, b, (short)0, c,
                                              false, false);
  }
  // C/D 16x16 f32: VGPR r -> row r (lanes0-15) / row r+8 (lanes16-31), col = lo
  float* yo = Y + (size_t)(m0 + hi * 8) * D + n0 + lo;
#pragma unroll
  for (int r = 0; r < 8; ++r) yo[(size_t)r * D] = c[r];
}

// ---------- al_s = xl@a_src, al_d = xl@a_dst : one wave per node ----------
__global__ void k_rowdot2(const float* __restrict__ xl, const float* __restrict__ a_s,
                          const float* __restrict__ a_d, float* __restrict__ al_s,
                          float* __restrict__ al_d, int N) {
  int n = blockIdx.x * (blockDim.x >> 5) + (threadIdx.x >> 5);
  if (n >= N) return;
  int lane = threadIdx.x & 31;
  float s1 = 0.f, s2 = 0.f;
#pragma unroll
  for (int j = 0; j < 4; ++j) {
    int f = lane + j * 32;
    float v = xl[(size_t)n * D + f];
    s1 += v * a_s[f];
    s2 += v * a_d[f];
  }
  s1 = wave_sum(s1);
  s2 = wave_sum(s2);
  if (lane == 0) { al_s[n] = s1; al_d[n] = s2; }
}

// ---------- e = leaky_relu(...); segment max via ordered-uint atomicMax ----------
__global__ void k_edge_score(const int* __restrict__ src, const int* __restrict__ dst,
                             const float* __restrict__ al_s, const float* __restrict__ al_d,
                             const float* __restrict__ eE, float* __restrict__ esc,
                             unsigned* __restrict__ seg_m, int E) {
  int k = blockIdx.x * blockDim.x + threadIdx.x;
  if (k >= E) return;
  int s = src[k], d = dst[k];
  float e = al_s[s] + al_d[d] + eE[k];
  e = (e > 0.f) ? e : 0.2f * e;
  esc[k] = e;
  atomicMax(seg_m + d, f32_ord(e));
}

// ---------- ex = exp(e - m[dst]); segment sum ----------
__global__ void k_edge_exp(const int* __restrict__ dst, float* __restrict__ esc,
                           const unsigned* __restrict__ seg_m, float* __restrict__ seg_s,
                           int E) {
  int k = blockIdx.x * blockDim.x + threadIdx.x;
  if (k >= E) return;
  int d = dst[k];
  float ex = expf(esc[k] - ord_f32(seg_m[d]));
  esc[k] = ex;
  atomicAdd(seg_s + d, ex);
}

// ---------- acc[dst] += alpha * xl[src] : one wave per edge ----------
__global__ void k_edge_scatter(const int* __restrict__ src, const int* __restrict__ dst,
                               const float* __restrict__ esc, const float* __restrict__ seg_s,
                               const float* __restrict__ xl, float* __restrict__ acc, int E) {
  int e = blockIdx.x * (blockDim.x >> 5) + (threadIdx.x >> 5);
  if (e >= E) return;
  int lane = threadIdx.x & 31;
  int s = src[e], d = dst[e];
  float alpha = esc[e] / (seg_s[d] + 1e-16f);
  const float* xs = xl + (size_t)s * D;
  float* ad = acc + (size_t)d * D;
#pragma unroll
  for (int j = 0; j < 4; ++j) {
    int f = lane + j * 32;
    atomicAdd(ad + f, xs[f] * alpha);
  }
}

// ---------- out = LN(acc + bias) * g + be : one wave per node ----------
__global__ void k_ln_bias(const float* __restrict__ acc, const float* __restrict__ bias,
                          const float* __restrict__ gw, const float* __restrict__ bw,
                          float* __restrict__ outp, int N) {
  int n = blockIdx.x * (blockDim.x >> 5) + (threadIdx.x >> 5);
  if (n >= N) return;
  int lane = threadIdx.x & 31;
  float v[4], s = 0.f, s2 = 0.f;
#pragma unroll
  for (int j = 0; j < 4; ++j) {
    int f = lane + j * 32;
    v[j] = acc[(size_t)n * D + f] + bias[f];
    s += v[j];
    s2 += v[j] * v[j];
  }
  s = wave_sum(s);
  s2 = wave_sum(s2);
  float mu = s * (1.0f / D);
  float var = s2 * (1.0f / D) - mu * mu;
  float rs = rsqrtf(var + 1e-5f);
#pragma unroll
  for (int j = 0; j < 4; ++j) {
    int f = lane + j * 32;
    outp[(size_t)n * D + f] = (v[j] - mu) * rs * gw[f] + bw[f];
  }
}

// ---------- global mean-pool accumulation : one wave per node ----------
__global__ void k_pool(const float* __restrict__ x4, const int* __restrict__ batch,
                       float* __restrict__ pooled, float* __restrict__ cnt, int N) {
  int n = blockIdx.x * (blockDim.x >> 5) + (threadIdx.x >> 5);
  if (n >= N) return;
  int lane = threadIdx.x & 31;
  int b = batch[n];
#pragma unroll
  for (int j = 0; j < 4; ++j) {
    int f = lane + j * 32;
    atomicAdd(pooled + (size_t)b * D + f, x4[(size_t)n * D + f]);
  }
  if (lane == 0) atomicAdd(cnt + b, 1.0f);
}

// ---------- out = relu(LN(mean @ Wl + bl)) : one block per graph ----------
__global__ void k_final(const float* __restrict__ pooled, const float* __restrict__ cnt,
                        const float* __restrict__ Wl, const float* __restrict__ bl,
                        const float* __restrict__ gl, const float* __restrict__ bel,
                        float* __restrict__ out) {
  __shared__ float mean_s[D];
  __shared__ float r1[D], r2[D];
  int g = blockIdx.x, t = threadIdx.x;
  float c = fmaxf(cnt[g], 1.0f);
  mean_s[t] = pooled[(size_t)g * D + t] / c;
  __syncthreads();
  float y = bl[t];
  for (int k = 0; k < D; ++k) y += mean_s[k] * Wl[(size_t)k * D + t];
  r1[t] = y;
  r2[t] = y * y;
  __syncthreads();
  for (int off = D / 2; off > 0; off >>= 1) {
    if (t < off) { r1[t] += r1[t + off]; r2[t] += r2[t + off]; }
    __syncthreads();
  }
  float mu = r1[0] * (1.0f / D);
  float var = r2[0] * (1.0f / D) - mu * mu;
  float rs = rsqrtf(var + 1e-5f);
  float o = (y - mu) * rs * gl[t] + bel[t];
  out[(size_t)g * D + t] = fmaxf(o, 0.0f);
}

extern "C" void kernel_launch(void* const* d_in, const int* in_sizes, int n_in,
                              void* d_out, int out_size, void* d_ws, size_t ws_size,
                              hipStream_t stream) {
  const float* x   = (const float*)d_in[0];
  const float* ea  = (const float*)d_in[1];
  const float* W1  = (const float*)d_in[2];
  const float* as1 = (const float*)d_in[3];
  const float* ad1 = (const float*)d_in[4];
  const float* We1 = (const float*)d_in[5];
  const float* ae1 = (const float*)d_in[6];
  const float* b1  = (const float*)d_in[7];
  const float* g1  = (const float*)d_in[8];
  const float* be1 = (const float*)d_in[9];
  const float* W2  = (const float*)d_in[10];
  const float* as2 = (const float*)d_in[11];
  const float* ad2 = (const float*)d_in[12];
  const float* We2 = (const float*)d_in[13];
  const float* ae2 = (const float*)d_in[14];
  const float* b2  = (const float*)d_in[15];
  const float* g2  = (const float*)d_in[16];
  const float* be2 = (const float*)d_in[17];
  const float* Wl  = (const float*)d_in[18];
  const float* bl  = (const float*)d_in[19];
  const float* gl  = (const float*)d_in[20];
  const float* bel = (const float*)d_in[21];
  const int* eidx  = (const int*)d_in[22];
  const int* batch = (const int*)d_in[23];
  float* out = (float*)d_out;

  const int N = in_sizes[0] / D;    // 50000 (multiple of 16)
  const int E = in_sizes[22] / 2;   // 600000
  const int G = out_size / D;       // 64
  const int* src = eidx;
  const int* dst = eidx + E;

  // ---- workspace layout (floats). acc/seg_m/seg_s contiguous for one-shot zeroing.
  float* ws = (float*)d_ws;
  size_t o = 0;
  float* xl    = ws + o; o += (size_t)N * D;
  float* xbuf  = ws + o; o += (size_t)N * D;   // x2 / x4
  float* acc   = ws + o; o += (size_t)N * D;   // attention output accumulator
  unsigned* seg_m = (unsigned*)(ws + o); o += N;
  float* seg_s = ws + o; o += N;
  float* al_s  = ws + o; o += N;
  float* al_d  = ws + o; o += N;
  float* eE1   = ws + o; o += E;
  float* eE2   = ws + o; o += E;
  float* esc   = ws + o; o += E;
  float* ce    = ws + o; o += 64;
  float* pooled= ws + o; o += (size_t)G * D;
  float* cnt   = ws + o; o += G;
  (void)ws_size; (void)n_in;

  const int wpb = 8;  // waves per 256-thread block
  dim3 blk(256);
  int nb_node = (N + wpb - 1) / wpb;
  int nb_edge_w = (E + wpb - 1) / wpb;
  int nb_edge_t = (E + 255) / 256;

  // edge-attr contribution, shared by both layers
  k_prep_ce<<<1, 64, 0, stream>>>(We1, ae1, We2, ae2, ce);
  k_edge_pre<<<nb_edge_w, blk, 0, stream>>>(ea, ce, eE1, eE2, E);

  const float* inX[2]  = {x, xbuf};
  const float* Ws[2]   = {W1, W2};
  const float* asv[2]  = {as1, as2};
  const float* adv[2]  = {ad1, ad2};
  const float* eEv[2]  = {eE1, eE2};
  const float* bv[2]   = {b1, b2};
  const float* gv[2]   = {g1, g2};
  const float* bev[2]  = {be1, be2};

  for (int L = 0; L < 2; ++L) {
    // zero acc + seg_m + seg_s (contiguous span)
    k_zero<<<2048, blk, 0, stream>>>(acc, (size_t)N * D + 2 * (size_t)N);
    k_gemm_xW<<<N / 16, blk, 0, stream>>>(inX[L], Ws[L], xl);
    k_rowdot2<<<nb_node, blk, 0, stream>>>(xl, asv[L], adv[L], al_s, al_d, N);
    k_edge_score<<<nb_edge_t, blk, 0, stream>>>(src, dst, al_s, al_d, eEv[L], esc, seg_m, E);
    k_edge_exp<<<nb_edge_t, blk, 0, stream>>>(dst, esc, seg_m, seg_s, E);
    k_edge_scatter<<<nb_edge_w, blk, 0, stream>>>(src, dst, esc, seg_s, xl, acc, E);
    k_ln_bias<<<nb_node, blk, 0, stream>>>(acc, bv[L], gv[L], bev[L], xbuf, N);
  }

  // pooling + head
  k_zero<<<64, blk, 0, stream>>>(pooled, (size_t)G * D + G);
  k_pool<<<nb_node, blk, 0, stream>>>(xbuf, batch, pooled, cnt, N);
  k_final<<<G, D, 0, stream>>>(pooled, cnt, Wl, bl, gl, bel, out);
}